// MultiHeadGATLayer_45028437131378
// MI455X (gfx1250) — compile-verified
//
#include <hip/hip_runtime.h>
#include <cmath>

#define F_IN   256
#define HID    256
#define HEADS  8
#define HD     32
#define SLOPE  0.2f
#define LN_EPS 1e-5f

typedef __attribute__((ext_vector_type(2))) float v2f;
typedef __attribute__((ext_vector_type(8))) float v8f;

// ---- order-preserving float<->uint key for atomicMax-based segment_max ----
__device__ __forceinline__ unsigned fkey(float f) {
    int b = __float_as_int(f);
    return (unsigned)(b >= 0 ? (b ^ 0x80000000) : ~b);
}
__device__ __forceinline__ float funkey(unsigned u) {
    int b = (u & 0x80000000u) ? (int)(u ^ 0x80000000u) : ~(int)u;
    return __int_as_float(b);
}

// ---------------------------------------------------------------------------
// Kernel 0: zero the accumulators (agg, den, mkey). Harness does not re-zero
// between graph replays, so this must run every call.
// ---------------------------------------------------------------------------
__global__ void k_zero(float* __restrict__ agg, float* __restrict__ den,
                       unsigned* __restrict__ mkey, size_t nAgg, size_t nDN) {
    size_t i = (size_t)blockIdx.x * blockDim.x + threadIdx.x;
    if (i < nAgg) {
        agg[i] = 0.0f;
    } else if (i < nAgg + nDN) {
        size_t j = i - nAgg;
        den[j]  = 0.0f;
        mkey[j] = 0u;            // fkey of anything finite beats 0
    }
}

// ---------------------------------------------------------------------------
// Kernel 1: H[n, h*32+d] = sum_k X[n,k] * W[h,k,d] + bw[h,d]
// fp32 WMMA 16x16x4, one wave per 16x16 output tile.
// Rows are CLAMPED (not predicated): unconditional clause-able loads in the
// K loop. Epilogue: uniform full-tile fast path -> unconditional stores.
// A layout: lanes 0-15 -> rows, K = k..k+1 ; lanes 16-31 -> rows, K = k+2..k+3
// B layout mirrors over columns. C/D: vgpr r -> row r + 8*(lane>=16).
// ---------------------------------------------------------------------------
__global__ void k_gemm_xw(const float* __restrict__ X, const float* __restrict__ W,
                          const float* __restrict__ bw, float* __restrict__ Hf, int N) {
    const int lane = threadIdx.x;
    const int hi   = lane >> 4;        // 0 or 1
    const int lo   = lane & 15;
    const int n0   = blockIdx.x * 16;
    const int c0   = blockIdx.y * 16;  // global output column tile (0..255)
    const int row  = min(n0 + lo, N - 1);   // clamp: loads always in-bounds
    const int col  = c0 + lo;
    const int head = col >> 5;
    const int d    = col & 31;
    // B[k][col] = W[head*F_IN*HD + k*HD + d]
    const float* Bp = W + (size_t)head * F_IN * HD + d;
    const float* Ap = X + (size_t)row * F_IN + 2 * hi;   // lane's K phase

    v8f acc = {};
#pragma unroll 4
    for (int k = 0; k < F_IN; k += 4) {
        const int kb = k + 2 * hi;
        const v2f a = *(const v2f*)(Ap + k);             // global_load_b64
        v2f b;
        b.x = Bp[(size_t)kb * HD];
        b.y = Bp[(size_t)(kb + 1) * HD];
        acc = __builtin_amdgcn_wmma_f32_16x16x4_f32(
            false, a, false, b, (short)0, acc, false, false);
    }
    const float bias = bw[col];   // bw flat: [HEADS*HD] == [256]
    float* outp = Hf + (size_t)(n0 + 8 * hi) * HID + col;
    if (n0 + 16 <= N) {           // uniform branch: full tile (always, N%16==0)
#pragma unroll
        for (int r = 0; r < 8; ++r) outp[(size_t)r * HID] = acc[r] + bias;
    } else {
#pragma unroll
        for (int r = 0; r < 8; ++r) {
            const int orow = n0 + r + 8 * hi;
            if (orow < N) Hf[(size_t)orow * HID + col] = acc[r] + bias;
        }
    }
}

// ---------------------------------------------------------------------------
// Kernel 2: si[n,h] = H[n,h,:] . A[h,:HD] ; sj[n,h] = H[n,h,:] . A[h,HD:]
// one thread per (node, head)
// ---------------------------------------------------------------------------
__global__ void k_sisj(const float* __restrict__ Hf, const float* __restrict__ A,
                       float* __restrict__ si, float* __restrict__ sj, int N) {
    const int t = blockIdx.x * blockDim.x + threadIdx.x;
    const int n = t >> 3, h = t & 7;
    if (n >= N) return;
    const float* hp = Hf + (size_t)n * HID + h * HD;
    const float* a1 = A + h * (2 * HD);
    const float* a2 = a1 + HD;
    float s1 = 0.0f, s2 = 0.0f;
#pragma unroll
    for (int d = 0; d < HD; d += 4) {
        float4 hv = *(const float4*)(hp + d);
        float4 x1 = *(const float4*)(a1 + d);
        float4 x2 = *(const float4*)(a2 + d);
        s1 += hv.x * x1.x + hv.y * x1.y + hv.z * x1.z + hv.w * x1.w;
        s2 += hv.x * x2.x + hv.y * x2.y + hv.z * x2.z + hv.w * x2.w;
    }
    si[t] = s1;
    sj[t] = s2;
}

// ---------------------------------------------------------------------------
// Kernel 3: per-edge scores e = leaky_relu(si[tgt]+sj[src]+ba), segment max.
// ---------------------------------------------------------------------------
__global__ void k_edge1(const int* __restrict__ tgt, const int* __restrict__ src,
                        const float* __restrict__ si, const float* __restrict__ sj,
                        const float* __restrict__ ba, float* __restrict__ ebuf,
                        unsigned* __restrict__ mkey, int E) {
    const int e = blockIdx.x * blockDim.x + threadIdx.x;
    if (e >= E) return;
    const int t = tgt[e], s = src[e];
    float sv[8], jv[8];
    *(float4*)(sv)     = *(const float4*)(si + (size_t)t * 8);
    *(float4*)(sv + 4) = *(const float4*)(si + (size_t)t * 8 + 4);
    *(float4*)(jv)     = *(const float4*)(sj + (size_t)s * 8);
    *(float4*)(jv + 4) = *(const float4*)(sj + (size_t)s * 8 + 4);
#pragma unroll
    for (int h = 0; h < HEADS; ++h) {
        float v = sv[h] + jv[h] + ba[h];
        v = (v >= 0.0f) ? v : SLOPE * v;
        ebuf[(size_t)e * HEADS + h] = v;
        atomicMax(&mkey[(size_t)t * HEADS + h], fkey(v));
    }
}

// ---------------------------------------------------------------------------
// Kernel 4: ex = exp(e - m[tgt]); ebuf <- ex; den[tgt] += ex
// ---------------------------------------------------------------------------
__global__ void k_edge2(const int* __restrict__ tgt, const unsigned* __restrict__ mkey,
                        float* __restrict__ ebuf, float* __restrict__ den, int E) {
    const int e = blockIdx.x * blockDim.x + threadIdx.x;
    if (e >= E) return;
    const int t = tgt[e];
#pragma unroll
    for (int h = 0; h < HEADS; ++h) {
        const float m  = funkey(mkey[(size_t)t * HEADS + h]);
        const float ex = __expf(ebuf[(size_t)e * HEADS + h] - m);
        ebuf[(size_t)e * HEADS + h] = ex;
        atomicAdd(&den[(size_t)t * HEADS + h], ex);
    }
}

// ---------------------------------------------------------------------------
// Kernel 5: agg[tgt, h*32+d] += (ex/den[tgt,h]) * H[src, h*32+d]
// one 256-thread block per edge: tx = h*32+d. Coalesced gather of H[src,:]
// and coalesced atomic adds into agg[tgt,:] (both rows L2-resident).
// ---------------------------------------------------------------------------
__global__ void k_edge3(const int* __restrict__ tgt, const int* __restrict__ src,
                        const float* __restrict__ Hf, const float* __restrict__ ebuf,
                        const float* __restrict__ den, float* __restrict__ agg) {
    const int e  = blockIdx.x;
    const int tx = threadIdx.x;          // 0..255
    const int h  = tx >> 5;
    const int t  = tgt[e], s = src[e];
    const float ex    = ebuf[(size_t)e * HEADS + h];
    const float dn    = den[(size_t)t * HEADS + h];   // > 0: includes own ex
    const float alpha = ex / dn;
    atomicAdd(&agg[(size_t)t * HID + tx], alpha * Hf[(size_t)s * HID + tx]);
}

// ---------------------------------------------------------------------------
// Kernel 6: per node: out = ELU(agg + H); LayerNorm over 32 dims per head;
// mean over heads -> meanbuf[n, 0..31]. One wave32 per node, lane = d.
// ---------------------------------------------------------------------------
__global__ void k_node(const float* __restrict__ Hf, const float* __restrict__ agg,
                       const float* __restrict__ gamma, const float* __restrict__ beta,
                       float* __restrict__ meanbuf, int N) {
    const int lane = threadIdx.x & 31;
    const int wid  = threadIdx.x >> 5;
    const int n    = blockIdx.x * 8 + wid;
    if (n >= N) return;
    float acc = 0.0f;
#pragma unroll
    for (int h = 0; h < HEADS; ++h) {
        const size_t idx = (size_t)n * HID + h * HD + lane;
        float v = agg[idx] + Hf[idx];
        v = (v > 0.0f) ? v : expm1f(v);
        float s = v;
#pragma unroll
        for (int m = 16; m; m >>= 1) s += __shfl_xor(s, m, 32);
        const float mu = s * (1.0f / 32.0f);
        const float dv = v - mu;
        float sq = dv * dv;
#pragma unroll
        for (int m = 16; m; m >>= 1) sq += __shfl_xor(sq, m, 32);
        const float var = sq * (1.0f / 32.0f);
        const float nrm = dv * rsqrtf(var + LN_EPS) * gamma[h * HD + lane] + beta[h * HD + lane];
        acc += nrm * 0.125f;   // mean over 8 heads
    }
    meanbuf[(size_t)n * HD + lane] = acc;
}

// ---------------------------------------------------------------------------
// Kernel 7: out = ELU(meanbuf @ Wout + bout), fp32 WMMA, K = 32.
// Same clamped-row loads + uniform full-tile store fast path.
// ---------------------------------------------------------------------------
__global__ void k_gemm_out(const float* __restrict__ M, const float* __restrict__ Wout,
                           const float* __restrict__ bout, float* __restrict__ out, int N) {
    const int lane = threadIdx.x;
    const int hi   = lane >> 4;
    const int lo   = lane & 15;
    const int n0   = blockIdx.x * 16;
    const int c0   = blockIdx.y * 16;
    const int row  = min(n0 + lo, N - 1);   // clamp
    const int col  = c0 + lo;
    const float* Ap = M + (size_t)row * HD + 2 * hi;
    const float* Bp = Wout + col;

    v8f acc = {};
#pragma unroll
    for (int k = 0; k < HD; k += 4) {
        const int kb = k + 2 * hi;
        const v2f a = *(const v2f*)(Ap + k);            // global_load_b64
        v2f b;
        b.x = Bp[(size_t)kb * HID];
        b.y = Bp[(size_t)(kb + 1) * HID];
        acc = __builtin_amdgcn_wmma_f32_16x16x4_f32(
            false, a, false, b, (short)0, acc, false, false);
    }
    const float bias = bout[col];
    float* outp = out + (size_t)(n0 + 8 * hi) * HID + col;
    if (n0 + 16 <= N) {           // uniform branch: full tile
#pragma unroll
        for (int r = 0; r < 8; ++r) {
            float v = acc[r] + bias;
            v = (v > 0.0f) ? v : expm1f(v);
            outp[(size_t)r * HID] = v;
        }
    } else {
#pragma unroll
        for (int r = 0; r < 8; ++r) {
            const int orow = n0 + r + 8 * hi;
            if (orow < N) {
                float v = acc[r] + bias;
                v = (v > 0.0f) ? v : expm1f(v);
                out[(size_t)orow * HID + col] = v;
            }
        }
    }
}

// ---------------------------------------------------------------------------
extern "C" void kernel_launch(void* const* d_in, const int* in_sizes, int n_in,
                              void* d_out, int out_size, void* d_ws, size_t ws_size,
                              hipStream_t stream) {
    const float* X     = (const float*)d_in[0];
    const int*   EI    = (const int*)  d_in[1];
    const float* W     = (const float*)d_in[2];
    const float* bw    = (const float*)d_in[3];
    const float* A     = (const float*)d_in[4];
    const float* ba    = (const float*)d_in[5];
    const float* gamma = (const float*)d_in[6];
    const float* beta  = (const float*)d_in[7];
    const float* Wout  = (const float*)d_in[8];
    const float* bout  = (const float*)d_in[9];
    float* out = (float*)d_out;

    const int N = in_sizes[0] / F_IN;
    const int E = in_sizes[1] / 2;
    const int* tgt = EI;
    const int* src = EI + E;

    // workspace layout (floats)
    float* ws = (float*)d_ws;
    size_t off = 0;
    float*    Hf      = ws + off; off += (size_t)N * HID;     // 12.8M
    float*    si      = ws + off; off += (size_t)N * HEADS;
    float*    sj      = ws + off; off += (size_t)N * HEADS;
    float*    den     = ws + off; off += (size_t)N * HEADS;
    unsigned* mkey    = (unsigned*)(ws + off); off += (size_t)N * HEADS;
    float*    agg     = ws + off; off += (size_t)N * HID;     // 12.8M
    float*    meanbuf = ws + off; off += (size_t)N * HD;
    float*    ebuf    = ws + off; off += (size_t)E * HEADS;   // 12.8M

    const size_t nAgg = (size_t)N * HID;
    const size_t nDN  = (size_t)N * HEADS;

    {   // 0: zero accumulators
        const size_t total = nAgg + nDN;
        const int blocks = (int)((total + 255) / 256);
        k_zero<<<blocks, 256, 0, stream>>>(agg, den, mkey, nAgg, nDN);
    }
    {   // 1: H = X @ W + bw   (fp32 WMMA)
        dim3 grid((N + 15) / 16, HID / 16);
        k_gemm_xw<<<grid, 32, 0, stream>>>(X, W, bw, Hf, N);
    }
    {   // 2: si, sj
        const int blocks = (N * HEADS + 255) / 256;
        k_sisj<<<blocks, 256, 0, stream>>>(Hf, A, si, sj, N);
    }
    {   // 3: edge scores + segment max
        const int blocks = (E + 255) / 256;
        k_edge1<<<blocks, 256, 0, stream>>>(tgt, src, si, sj, ba, ebuf, mkey, E);
    }
    {   // 4: exp + segment sum
        const int blocks = (E + 255) / 256;
        k_edge2<<<blocks, 256, 0, stream>>>(tgt, mkey, ebuf, den, E);
    }
    {   // 5: weighted scatter-add of messages (one block per edge)
        k_edge3<<<E, 256, 0, stream>>>(tgt, src, Hf, ebuf, den, agg);
    }
    {   // 6: skip + ELU + LayerNorm + head mean
        const int blocks = (N + 7) / 8;
        k_node<<<blocks, 256, 0, stream>>>(Hf, agg, gamma, beta, meanbuf, N);
    }
    {   // 7: out = ELU(mean @ Wout + bout)   (fp32 WMMA)
        dim3 grid((N + 15) / 16, HID / 16);
        k_gemm_out<<<grid, 32, 0, stream>>>(meanbuf, Wout, bout, out, N);
    }
}